// CRF_54915451846732
// MI455X (gfx1250) — compile-verified
//
#include <hip/hip_runtime.h>

typedef __attribute__((ext_vector_type(2))) float v2f;
typedef __attribute__((ext_vector_type(8))) float v8f;

namespace {
constexpr int Tn = 512;
constexpr int Bn = 2048;
constexpr int Kn = 32;
constexpr int START = Kn - 3;   // 29
constexpr int END_S = Kn - 2;   // 30
}

#if defined(__has_builtin)
#if __has_builtin(__builtin_amdgcn_permlanex16)
#define XHALF_PERMLANE 1
#endif
#endif

// Exchange value with the corresponding lane of the other 16-lane half.
__device__ __forceinline__ float xhalf(float v) {
#ifdef XHALF_PERMLANE
  unsigned u = __float_as_uint(v);
  unsigned r = __builtin_amdgcn_permlanex16(u, u, 0x76543210u, 0xfedcba98u,
                                            false, false);
  return __uint_as_float(r);
#else
  return __shfl_xor(v, 16, 32);
#endif
}

// One wave32 per block. Wave owns batch rows [b0, b0+16).
// State: score[b,i] = mreg + log(E[b,i]); E kept in WMMA B-fragment layout
// (unnormalized; rescaled by max every 4 steps).
// Per step: accT(j,b) = sum_i P[j,i]*E[b,i] via V_WMMA_F32_16X16X4_F32.
__global__ __launch_bounds__(32)
void crf_forward_wmma(const float* __restrict__ hid,
                      const float* __restrict__ msk,
                      const float* __restrict__ trn,
                      float* __restrict__ out)
{
  const int lane = threadIdx.x & 31;
  const int bl   = lane & 15;   // batch row within tile (N index of D)
  const int h    = lane >> 4;   // lane half
  const int b0   = blockIdx.x * 16;

  // ---- global max of transition (exp-domain normalizer) ----
  float tmx = -3.4e38f;
  #pragma unroll
  for (int i = 0; i < Kn; ++i) tmx = fmaxf(tmx, trn[lane * Kn + i]);
  #pragma unroll
  for (int m = 16; m >= 1; m >>= 1) tmx = fmaxf(tmx, __shfl_xor(tmx, m, 32));

  // ---- A fragments: P = exp(trans - tmx), constant across t ----
  // A tile tl covers j = 16*tl + (lane&15); K index = 4c + 2h + v.
  v2f A[2][8];
  #pragma unroll
  for (int tl = 0; tl < 2; ++tl) {
    #pragma unroll
    for (int c = 0; c < 8; ++c) {
      const int j  = 16 * tl + bl;
      const int i0 = 4 * c + 2 * h;
      A[tl][c][0] = __expf(trn[j * Kn + i0]     - tmx);
      A[tl][c][1] = __expf(trn[j * Kn + i0 + 1] - tmx);
    }
  }

  // ---- initial state: score0 = NEG except START=0  ->  m=0, e=one-hot ----
  v2f E[8];
  #pragma unroll
  for (int c = 0; c < 8; ++c) {
    const int i0 = 4 * c + 2 * h;
    E[c][0] = (i0     == START) ? 1.0f : 0.0f;
    E[c][1] = (i0 + 1 == START) ? 1.0f : 0.0f;
  }
  float mreg = 0.0f;

  const float* hrow = hid + (size_t)(b0 + bl) * Kn + 8 * h;

  for (int t = 0; t < Tn; ++t) {
    // accT = P x E^T : two 16x16 f32 D tiles (j = 0..15 and 16..31)
    v8f d0 = {}; v8f d1 = {};
    #pragma unroll
    for (int c = 0; c < 8; ++c) {
      d0 = __builtin_amdgcn_wmma_f32_16x16x4_f32(false, A[0][c], false, E[c],
                                                 (short)0, d0, false, false);
      d1 = __builtin_amdgcn_wmma_f32_16x16x4_f32(false, A[1][c], false, E[c],
                                                 (short)0, d1, false, false);
    }

    // hidden tile: this lane holds (b = bl, j = 8h + r) / (j = 16 + 8h + r)
    const float* hp = hrow + (size_t)t * (Bn * Kn);
    const float4 ha = *(const float4*)(hp);
    const float4 hb = *(const float4*)(hp + 4);
    const float4 hc = *(const float4*)(hp + 16);
    const float4 hd = *(const float4*)(hp + 20);
    __builtin_prefetch(hp + Bn * Kn, 0, 3);   // next step's tile

    float hv[16] = {ha.x, ha.y, ha.z, ha.w, hb.x, hb.y, hb.z, hb.w,
                    hc.x, hc.y, hc.z, hc.w, hd.x, hd.y, hd.z, hd.w};

    // Hmax over j for this b: 16 in-lane values + one cross-half exchange
    float hm = hv[0];
    #pragma unroll
    for (int i = 1; i < 16; ++i) hm = fmaxf(hm, hv[i]);
    hm = fmaxf(hm, xhalf(hm));

    // u_j = acc_j * exp(hidden_j - Hmax)   (kept unnormalized)
    float u[16];
    #pragma unroll
    for (int i = 0; i < 8; ++i) u[i]     = d0[i] * __expf(hv[i]     - hm);
    #pragma unroll
    for (int i = 0; i < 8; ++i) u[i + 8] = d1[i] * __expf(hv[i + 8] - hm);

    const float newm = mreg + tmx + hm;     // no log on the per-step path

    // ---- D-layout -> B-fragment layout, paired: 8 cross-half exchanges ----
    // value (b,i) resides at lane half (i>>3)&1, reg idx(i) = (i>>4)*8 + (i&7);
    // idx() ignores bit3, so fragments c and c+2 share source registers and
    // one exchange serves both halves with a per-half payload select.
    v2f En[8];
    {
      const int c0list[4] = {0, 1, 4, 5};
      #pragma unroll
      for (int p = 0; p < 4; ++p) {
        const int c0 = c0list[p];
        #pragma unroll
        for (int v = 0; v < 2; ++v) {
          const int iA   = 4 * c0 + v;
          const int idxA = ((iA >> 4) * 8) + (iA & 7);
          const int idxB = (((iA + 2) >> 4) * 8) + ((iA + 2) & 7);
          const float payload = h ? u[idxA] : u[idxB];
          const float sh = xhalf(payload);
          En[c0][v]     = h ? sh       : u[idxA];
          En[c0 + 2][v] = h ? u[idxB]  : sh;
        }
      }
    }

    // ---- per-row mask select (b uniform per lane -> plain cndmask) ----
    const float mk  = msk[t * Bn + b0 + bl];
    const bool keep = mk > 0.5f;
    #pragma unroll
    for (int c = 0; c < 8; ++c) {
      E[c][0] = keep ? E[c][0] : En[c][0];
      E[c][1] = keep ? E[c][1] : En[c][1];
    }
    mreg = keep ? mreg : newm;

    // ---- periodic renormalization (representation-neutral) ----
    if ((t & 3) == 3) {
      float U = E[0][0];
      #pragma unroll
      for (int c = 0; c < 8; ++c) U = fmaxf(U, fmaxf(E[c][0], E[c][1]));
      U = fmaxf(fmaxf(U, xhalf(U)), 1e-38f);
      const float rcpU = 1.0f / U;
      #pragma unroll
      for (int c = 0; c < 8; ++c) { E[c][0] *= rcpU; E[c][1] *= rcpU; }
      mreg += __logf(U);
    }

    if (__all((int)keep)) break;   // masks monotone in t: state is final
  }

  // ---- final renorm then out[b] = mreg + qmax + log(sum_i E*exp(tq-qmax)) ----
  {
    float U = E[0][0];
    #pragma unroll
    for (int c = 0; c < 8; ++c) U = fmaxf(U, fmaxf(E[c][0], E[c][1]));
    U = fmaxf(fmaxf(U, xhalf(U)), 1e-38f);
    const float rcpU = 1.0f / U;
    #pragma unroll
    for (int c = 0; c < 8; ++c) { E[c][0] *= rcpU; E[c][1] *= rcpU; }
    mreg += __logf(U);
  }

  float tq[8][2];
  float qm = -3.4e38f;
  #pragma unroll
  for (int c = 0; c < 8; ++c) {
    const int i0 = 4 * c + 2 * h;
    tq[c][0] = trn[END_S * Kn + i0];
    tq[c][1] = trn[END_S * Kn + i0 + 1];
    qm = fmaxf(qm, fmaxf(tq[c][0], tq[c][1]));
  }
  qm = fmaxf(qm, xhalf(qm));
  float s = 0.0f;
  #pragma unroll
  for (int c = 0; c < 8; ++c)
    s += E[c][0] * __expf(tq[c][0] - qm) + E[c][1] * __expf(tq[c][1] - qm);
  s += xhalf(s);
  if (h == 0) out[b0 + bl] = mreg + qm + __logf(s);
}

extern "C" void kernel_launch(void* const* d_in, const int* in_sizes, int n_in,
                              void* d_out, int out_size, void* d_ws, size_t ws_size,
                              hipStream_t stream)
{
  (void)in_sizes; (void)n_in; (void)d_ws; (void)ws_size; (void)out_size;
  const float* hid = (const float*)d_in[0];   // (T, B, K) f32
  const float* msk = (const float*)d_in[1];   // (T, B)    f32
  const float* trn = (const float*)d_in[2];   // (K, K)    f32
  float* out = (float*)d_out;                 // (B,)      f32
  crf_forward_wmma<<<Bn / 16, 32, 0, stream>>>(hid, msk, trn, out);
}